// ODERNN_Time_4200478015953
// MI455X (gfx1250) — compile-verified
//
#include <hip/hip_runtime.h>
#include <hip/hip_bf16.h>
#include <cstddef>

// ---------------------------------------------------------------------------
// ODE-RNN: kernel 1 = serial Euler ODE scan (produces per-batch-row initial
// GRU state), kernel 2 = fused GRU recurrence on the WMMA fp32 pipe.
// ---------------------------------------------------------------------------

typedef __attribute__((ext_vector_type(2))) float v2f;
typedef __attribute__((ext_vector_type(4))) float v4f;
typedef __attribute__((ext_vector_type(8))) float v8f;

#define BATCH   4096
#define SEQ     512
#define DIN     32
#define HID     64
#define ODEW    50

// V_WMMA_F32_16X16X4_F32: D(16x16 f32) = A(16x4 f32) * B(4x16 f32) + C
__device__ __forceinline__ v8f wmma4(v2f a, v2f b, v8f c) {
  return __builtin_amdgcn_wmma_f32_16x16x4_f32(
      /*neg_a=*/false, a, /*neg_b=*/false, b,
      /*c_mod=*/(short)0, c, /*reuse_a=*/false, /*reuse_b=*/false);
}

__device__ __forceinline__ float sigmoidf_(float x) {
  return 1.0f / (1.0f + __expf(-x));
}

// ---------------------------------------------------------------------------
// Kernel 1: serial Euler integration of the neural ODE.
// traj[k] = h BEFORE step k (matches lax.scan output), h0 = 0.
// One workgroup of 64 threads (thread j owns hidden unit j).
// ---------------------------------------------------------------------------
__global__ void __launch_bounds__(64) ode_traj_kernel(
    const float* __restrict__ t,
    const float* __restrict__ w1, const float* __restrict__ b1,
    const float* __restrict__ w2, const float* __restrict__ b2,
    float* __restrict__ traj)
{
  __shared__ float sw1[ODEW * HID];   // (50,64) row-major
  __shared__ float sw2[HID * ODEW];   // (64,50) row-major
  __shared__ float sb1[ODEW];
  __shared__ float sb2[HID];
  __shared__ float sh[HID];
  __shared__ float sz[ODEW];

  const int j = threadIdx.x;
  for (int i = j; i < ODEW * HID; i += 64) sw1[i] = w1[i];
  for (int i = j; i < HID * ODEW; i += 64) sw2[i] = w2[i];
  if (j < ODEW) sb1[j] = b1[j];
  sb2[j] = b2[j];

  float hj = 0.0f;
  __syncthreads();

  for (int k = 0; k < BATCH; ++k) {
    sh[j] = hj;
    __syncthreads();                       // sh visible
    traj[(size_t)k * HID + j] = hj;        // state BEFORE update

    if (j < ODEW) {
      float a = sb1[j];
      #pragma unroll 8
      for (int i = 0; i < HID; ++i) a += sh[i] * sw1[j * HID + i];
      sz[j] = tanhf(a);
    }
    __syncthreads();                       // sz visible

    const float dtk = t[k + 1] - t[k];
    float a2 = sb2[j];
    #pragma unroll 5
    for (int i = 0; i < ODEW; ++i) a2 += sz[i] * sw2[j * ODEW + i];
    hj += dtk * a2;
    __syncthreads();                       // all sz/sh reads done before rewrite
  }
}

// ---------------------------------------------------------------------------
// Kernel 2: fused GRU over 512 steps on the fp32 WMMA pipe.
// One wave (32 threads) per block; each wave owns a 16-row batch tile for all
// 512 steps. Weight B-fragments are step-invariant -> the compiler hoists them
// into the extended (>256) VGPR file. The x tile is software-pipelined:
// global->regs for step s+1 issued before the WMMAs of step s, regs->LDS
// (ping-pong buffer) after them. Single wave => in-order DS, no barriers.
// ---------------------------------------------------------------------------
__global__ void __launch_bounds__(32) gru_wmma_kernel(
    const float* __restrict__ x,     // (4096, 512, 32)
    const float* __restrict__ traj,  // (4096, 64) initial h per batch row
    const float* __restrict__ Wih,   // (192, 32)
    const float* __restrict__ Whh,   // (192, 64)
    const float* __restrict__ bih,   // (192,)
    const float* __restrict__ bhh,   // (192,)
    const float* __restrict__ Wfc,   // (1, 64)
    const float* __restrict__ bfc,   // (1,)
    float* __restrict__ out)         // (4096, 512)
{
  __shared__ __align__(16) float sX[2][16 * DIN];  // ping-pong x tile
  __shared__ __align__(16) float sH[16 * HID];     // h tile (row-major)
  __shared__ __align__(16) float sWfc[HID];

  const int lane = threadIdx.x & 31;
  const int lo   = lane & 15;     // col/row-within-16 index
  const int hi   = lane >> 4;     // half-wave select
  const int r0   = blockIdx.x * 16;   // first batch row of this tile

  sWfc[lane]      = Wfc[lane];
  sWfc[lane + 32] = Wfc[lane + 32];
  const float bfc0 = bfc[0];

  // Per-lane bias registers (same value for every row of a tile column).
  float brz[8], bxn[4], bhn[4];
  #pragma unroll
  for (int tt = 0; tt < 8; ++tt) brz[tt] = bih[tt * 16 + lo] + bhh[tt * 16 + lo];
  #pragma unroll
  for (int tt = 0; tt < 4; ++tt) {
    bxn[tt] = bih[128 + tt * 16 + lo];
    bhn[tt] = bhh[128 + tt * 16 + lo];
  }

  // Initial h: row-major copy into sH (A-fragment source) ...
  {
    v4f* dst = (v4f*)&sH[0];
    const v4f* src = (const v4f*)&traj[(size_t)r0 * HID];
    #pragma unroll
    for (int i = 0; i < 8; ++i) dst[lane + 32 * i] = src[lane + 32 * i];
  }
  // ... and D-tile-layout copy in registers (for the z*h_old blend).
  v8f hD[4];
  #pragma unroll
  for (int tt = 0; tt < 4; ++tt)
    #pragma unroll
    for (int v = 0; v < 8; ++v)
      hD[tt][v] = traj[(size_t)(r0 + v + 8 * hi) * HID + tt * 16 + lo];

  // x-tile pipeline helpers: 128 float4 slots per tile, 4 per lane.
  v4f xr[4];
  auto load_x_regs = [&](int s) {
    #pragma unroll
    for (int i = 0; i < 4; ++i) {
      const int f   = lane + 32 * i;
      const int row = f >> 3;
      const int q   = f & 7;
      xr[i] = ((const v4f*)&x[((size_t)(r0 + row) * SEQ + s) * DIN])[q];
    }
  };
  auto store_x_lds = [&](int buf) {
    v4f* dx = (v4f*)&sX[buf][0];
    #pragma unroll
    for (int i = 0; i < 4; ++i) dx[lane + 32 * i] = xr[i];
  };

  load_x_regs(0);
  store_x_lds(0);

  for (int s = 0; s < SEQ; ++s) {
    const int cur = s & 1;

    // Pipeline: issue next tile's global loads now; consume them at the
    // bottom of this step. Deep prefetch pulls lines into L2/WGP$ early.
    if (s + 1 < SEQ) load_x_regs(s + 1);
    if (s + 8 < SEQ) {
      const char* p = (const char*)
          &x[((size_t)(r0 + (lane >> 1)) * SEQ + (s + 8)) * DIN] + (lane & 1) * 64;
      __builtin_prefetch(p, 0, 1);   // global_prefetch_b8
    }

    // ---- accumulators: 8 tiles r|z (x+h fused), 4 xn, 4 hn ----
    v8f accRZ[8], accXN[4], accHN[4];
    #pragma unroll
    for (int tt = 0; tt < 8; ++tt) {
      v8f a;
      #pragma unroll
      for (int q = 0; q < 8; ++q) a[q] = brz[tt];
      accRZ[tt] = a;
    }
    #pragma unroll
    for (int tt = 0; tt < 4; ++tt) {
      v8f a, b;
      #pragma unroll
      for (int q = 0; q < 8; ++q) { a[q] = bxn[tt]; b[q] = bhn[tt]; }
      accXN[tt] = a;
      accHN[tt] = b;
    }

    // ---- x @ Wih.T : 8 K-chunks x 12 N-tiles of v_wmma_f32_16x16x4_f32 ----
    #pragma unroll
    for (int k = 0; k < 8; ++k) {
      const v2f ax = *(const v2f*)&sX[cur][lo * DIN + k * 4 + 2 * hi];
      #pragma unroll
      for (int tt = 0; tt < 12; ++tt) {
        const v2f bw = *(const v2f*)&Wih[(size_t)(tt * 16 + lo) * DIN + k * 4 + 2 * hi];
        if (tt < 8) accRZ[tt]     = wmma4(ax, bw, accRZ[tt]);
        else        accXN[tt - 8] = wmma4(ax, bw, accXN[tt - 8]);
      }
    }

    // ---- h @ Whh.T : 16 K-chunks x 12 N-tiles ----
    #pragma unroll
    for (int k = 0; k < 16; ++k) {
      const v2f ah = *(const v2f*)&sH[lo * HID + k * 4 + 2 * hi];
      #pragma unroll
      for (int tt = 0; tt < 12; ++tt) {
        const v2f bw = *(const v2f*)&Whh[(size_t)(tt * 16 + lo) * HID + k * 4 + 2 * hi];
        if (tt < 8) accRZ[tt]     = wmma4(ah, bw, accRZ[tt]);
        else        accHN[tt - 8] = wmma4(ah, bw, accHN[tt - 8]);
      }
    }

    // Drain the pipelined global loads into the other LDS buffer. DS ops are
    // in-order within a wave, so this cannot pass the A-fragment loads above.
    if (s + 1 < SEQ) store_x_lds(cur ^ 1);

    // ---- gates + state update (D-tile layout: row v+8*hi, col tt*16+lo) ----
    #pragma unroll
    for (int tt = 0; tt < 4; ++tt) {
      #pragma unroll
      for (int v = 0; v < 8; ++v) {
        const float rg = sigmoidf_(accRZ[tt][v]);
        const float zg = sigmoidf_(accRZ[4 + tt][v]);
        const float ng = tanhf(accXN[tt][v] + rg * accHN[tt][v]);
        const float hnew = (1.0f - zg) * ng + zg * hD[tt][v];
        hD[tt][v] = hnew;
        sH[(v + 8 * hi) * HID + tt * 16 + lo] = hnew;  // A-source for next step
      }
    }

    // ---- fc head: out[b][s] = h_new . Wfc + bfc (16 rows / wave) ----
    if (lane < 16) {
      const v4f* hr = (const v4f*)&sH[lane * HID];
      const v4f* wf = (const v4f*)sWfc;
      float acc = bfc0;
      #pragma unroll
      for (int q = 0; q < 16; ++q) {
        const v4f hv = hr[q];
        const v4f wv = wf[q];
        acc += hv[0] * wv[0] + hv[1] * wv[1] + hv[2] * wv[2] + hv[3] * wv[3];
      }
      out[(size_t)(r0 + lane) * SEQ + s] = acc;
    }
  }
}

// ---------------------------------------------------------------------------
// Launch. Inputs (setup_inputs order):
// 0:x 1:t 2:w1 3:b1 4:w2 5:b2 6:W_ih 7:W_hh 8:b_ih 9:b_hh 10:W_fc 11:b_fc
// ---------------------------------------------------------------------------
extern "C" void kernel_launch(void* const* d_in, const int* in_sizes, int n_in,
                              void* d_out, int out_size, void* d_ws, size_t ws_size,
                              hipStream_t stream) {
  const float* x    = (const float*)d_in[0];
  const float* t    = (const float*)d_in[1];
  const float* w1   = (const float*)d_in[2];
  const float* b1   = (const float*)d_in[3];
  const float* w2   = (const float*)d_in[4];
  const float* b2   = (const float*)d_in[5];
  const float* Wih  = (const float*)d_in[6];
  const float* Whh  = (const float*)d_in[7];
  const float* bih  = (const float*)d_in[8];
  const float* bhh  = (const float*)d_in[9];
  const float* Wfc  = (const float*)d_in[10];
  const float* bfc  = (const float*)d_in[11];
  float* out  = (float*)d_out;
  float* traj = (float*)d_ws;               // 4096*64 f32 = 1 MB scratch

  ode_traj_kernel<<<1, 64, 0, stream>>>(t, w1, b1, w2, b2, traj);
  gru_wmma_kernel<<<BATCH / 16, 32, 0, stream>>>(
      x, traj, Wih, Whh, bih, bhh, Wfc, bfc, out);
}